// MoeModel_81226421502503
// MI455X (gfx1250) — compile-verified
//
#include <hip/hip_runtime.h>

// MoE token-routed model, MI455X (gfx1250, wave32).
// Strategy: fold w2 into expert/residual weights once per block, then per
// 16-token tile do D = A(16x16 h) x B(16x16 packed weights) with chained
// V_WMMA_F32_16X16X4_F32 (K=16 -> 4 chunks). 12 WMMAs / 16 tokens.
// Epilogue (softmax/argmax/select) via per-wave LDS transpose.

typedef __attribute__((ext_vector_type(2))) float v2f;
typedef __attribute__((ext_vector_type(8))) float v8f;

#define WAVES_PER_BLOCK 8
#define BLOCK (WAVES_PER_BLOCK * 32)

__global__ __launch_bounds__(BLOCK) void moe_wmma_kernel(
    const float* __restrict__ x,        // [ntok,4]
    const float* __restrict__ w1,       // [4,16]
    const float* __restrict__ b1,       // [16]
    const float* __restrict__ gate_w,   // [16,8]
    const float* __restrict__ expert_w, // [8,16,16]
    const float* __restrict__ expert_b, // [8,16]
    const float* __restrict__ res_w,    // [16,16]
    const float* __restrict__ res_b,    // [16]
    const float* __restrict__ comb_w,   // [16,2]
    const float* __restrict__ comb_b,   // [2]
    const float* __restrict__ w2,       // [16,4]
    const float* __restrict__ b2,       // [4]
    float* __restrict__ out,            // [ntok,4]
    int ntok)
{
    // ---- folded weights (block-shared) ----
    __shared__ float sW1[64];        // w1[d][f]
    __shared__ float sB1[16];
    __shared__ float sGate[16 * 8];  // gate_w[f][e]
    __shared__ float sComb[16 * 2];  // comb_w[f][c]
    __shared__ float sResF[16 * 4];  // (res_w @ w2)[f][c]
    __shared__ float sExF[8 * 16 * 4]; // (We @ w2)[e][f][c]
    __shared__ float sBeF[8 * 4];    // (be @ w2)[e][c]
    // ---- per-wave D-tile transpose regions ----
    __shared__ float sDgc[WAVES_PER_BLOCK][2][16][16]; // gate|comb|res' outputs
    __shared__ float sDex[WAVES_PER_BLOCK][2][16][32]; // 8 experts x 4 cols

    const int tid = threadIdx.x;

    for (int i = tid; i < 64; i += BLOCK) sW1[i] = w1[i];
    for (int i = tid; i < 16; i += BLOCK) sB1[i] = b1[i];
    for (int i = tid; i < 128; i += BLOCK) sGate[i] = gate_w[i];
    for (int i = tid; i < 32; i += BLOCK) sComb[i] = comb_w[i];
    for (int i = tid; i < 64; i += BLOCK) {           // res_w @ w2
        int f = i >> 2, c = i & 3;
        float acc = 0.f;
        for (int g = 0; g < 16; ++g) acc = fmaf(res_w[f * 16 + g], w2[g * 4 + c], acc);
        sResF[i] = acc;
    }
    for (int i = tid; i < 512; i += BLOCK) {          // We @ w2
        int e = i >> 6, f = (i >> 2) & 15, c = i & 3;
        float acc = 0.f;
        for (int g = 0; g < 16; ++g) acc = fmaf(expert_w[(e * 16 + f) * 16 + g], w2[g * 4 + c], acc);
        sExF[i] = acc;
    }
    for (int i = tid; i < 32; i += BLOCK) {           // be @ w2
        int e = i >> 2, c = i & 3;
        float acc = 0.f;
        for (int g = 0; g < 16; ++g) acc = fmaf(expert_b[e * 16 + g], w2[g * 4 + c], acc);
        sBeF[i] = acc;
    }
    __syncthreads();

    const int lane = tid & 31;
    const int w    = tid >> 5;
    const int hi   = lane >> 4;   // which K-half this lane supplies
    const int ln   = lane & 15;   // row (A) / column (B,C,D) index

    // ---- loop-invariant per-lane registers ----
    // A-layout (f32 16x4): chunk k, vgpr j -> K/f = 4k + 2*hi + j.
    // B-layout assumed symmetric (matches ISA 8/16-bit B pattern):
    //   chunk k, vgpr j, lane -> B[K=4k+2*hi+j][N=ln].
    float wreg[4][2][4], breg[4][2];
    v2f bgc[4], bex0[4], bex1[4];
    for (int k = 0; k < 4; ++k) {
        for (int j = 0; j < 2; ++j) {
            const int f = 4 * k + 2 * hi + j;
            for (int d = 0; d < 4; ++d) wreg[k][j][d] = sW1[d * 16 + f];
            breg[k][j] = sB1[f];
            float gcv;
            if (ln < 8)       gcv = sGate[f * 8 + ln];          // gate logits: cols 0..7
            else if (ln < 10) gcv = sComb[f * 2 + (ln - 8)];    // comb logits: cols 8,9
            else if (ln < 12) gcv = 0.f;                        // pad
            else              gcv = sResF[f * 4 + (ln - 12)];   // res': cols 12..15
            bgc[k][j]  = gcv;
            bex0[k][j] = sExF[(ln >> 2) * 64 + f * 4 + (ln & 3)];        // experts 0..3
            bex1[k][j] = sExF[((ln >> 2) + 4) * 64 + f * 4 + (ln & 3)];  // experts 4..7
        }
    }
    float brF[4], b2r[4];
    for (int c = 0; c < 4; ++c) {
        float acc = 0.f;
        for (int g = 0; g < 16; ++g) acc = fmaf(res_b[g], w2[g * 4 + c], acc);
        brF[c] = acc;
        b2r[c] = b2[c];
    }
    const float dcb = comb_b[1] - comb_b[0];

    const float4* x4 = (const float4*)x;
    float4*       o4 = (float4*)out;

    const int nwaves = gridDim.x * WAVES_PER_BLOCK;
    const int gwave  = blockIdx.x * WAVES_PER_BLOCK + w;
    const int niter  = ntok >> 5;  // 32 tokens (2 WMMA tiles) per wave-iteration

    for (int it = gwave; it < niter; it += nwaves) {
        const int base = it << 5;
        v8f dgc[2], de0[2], de1[2];

        for (int t = 0; t < 2; ++t) {
            int tok = base + 16 * t + ln;
            if (tok >= ntok) tok = ntok - 1;
            const float4 xv = x4[tok];

            // h row directly in A layout: 8 values per lane (proj1 is K=4, VALU)
            v2f a[4];
            for (int k = 0; k < 4; ++k) {
                for (int j = 0; j < 2; ++j) {
                    float h = breg[k][j];
                    h = fmaf(xv.x, wreg[k][j][0], h);
                    h = fmaf(xv.y, wreg[k][j][1], h);
                    h = fmaf(xv.z, wreg[k][j][2], h);
                    h = fmaf(xv.w, wreg[k][j][3], h);
                    a[k][j] = h;
                }
            }

            v8f dg = {}; v8f d0 = {}; v8f d1 = {};
            for (int k = 0; k < 4; ++k) {
                dg = __builtin_amdgcn_wmma_f32_16x16x4_f32(false, a[k], false, bgc[k],
                                                           (short)0, dg, false, false);
                d0 = __builtin_amdgcn_wmma_f32_16x16x4_f32(false, a[k], false, bex0[k],
                                                           (short)0, d0, false, false);
                d1 = __builtin_amdgcn_wmma_f32_16x16x4_f32(false, a[k], false, bex1[k],
                                                           (short)0, d1, false, false);
            }
            dgc[t] = dg; de0[t] = d0; de1[t] = d1;
        }

        // D layout: vgpr r, lanes0-15 -> row r, lanes16-31 -> row r+8; col = ln.
        for (int t = 0; t < 2; ++t) {
            for (int r = 0; r < 8; ++r) {
                const int row = r + 8 * hi;
                sDgc[w][t][row][ln]      = dgc[t][r];
                sDex[w][t][row][ln]      = de0[t][r];
                sDex[w][t][row][16 + ln] = de1[t][r];
            }
        }
        asm volatile("" ::: "memory");
        __builtin_amdgcn_wave_barrier();   // same-wave DS ops are in-order in HW

        // epilogue: one token per lane (token = base + lane)
        {
            const int   t   = hi;
            const int   m   = ln;
            const float* row = &sDgc[w][t][m][0];

            float vmax = row[0]; int idx = 0;
            for (int e = 1; e < 8; ++e) {
                const float le = row[e];
                if (le > vmax) { vmax = le; idx = e; }
            }
            float s = 0.f;
            for (int e = 0; e < 8; ++e) s += __expf(row[e] - vmax);
            const float topp = __builtin_amdgcn_rcpf(s);                       // max prob
            const float c0 = __builtin_amdgcn_rcpf(1.f + __expf(row[9] - row[8] + dcb));
            const float c1 = 1.f - c0;
            const float sc = topp * c0;

            const float* exr = &sDex[w][t][m][idx * 4];
            const float* ber = &sBeF[idx * 4];

            float4 o;
            o.x = fmaf(exr[0] + ber[0], sc, fmaf(row[12] + brF[0], c1, b2r[0]));
            o.y = fmaf(exr[1] + ber[1], sc, fmaf(row[13] + brF[1], c1, b2r[1]));
            o.z = fmaf(exr[2] + ber[2], sc, fmaf(row[14] + brF[2], c1, b2r[2]));
            o.w = fmaf(exr[3] + ber[3], sc, fmaf(row[15] + brF[3], c1, b2r[3]));

            const int token = base + lane;
            if (token < ntok) o4[token] = o;
        }
        asm volatile("" ::: "memory");
        __builtin_amdgcn_wave_barrier();   // keep next-iter stores after these loads
    }
}

extern "C" void kernel_launch(void* const* d_in, const int* in_sizes, int n_in,
                              void* d_out, int out_size, void* d_ws, size_t ws_size,
                              hipStream_t stream) {
    const float* x        = (const float*)d_in[0];
    const float* w1       = (const float*)d_in[1];
    const float* b1       = (const float*)d_in[2];
    const float* gate_w   = (const float*)d_in[3];
    const float* expert_w = (const float*)d_in[4];
    const float* expert_b = (const float*)d_in[5];
    const float* res_w    = (const float*)d_in[6];
    const float* res_b    = (const float*)d_in[7];
    const float* comb_w   = (const float*)d_in[8];
    const float* comb_b   = (const float*)d_in[9];
    const float* w2       = (const float*)d_in[10];
    const float* b2       = (const float*)d_in[11];

    const int ntok = in_sizes[0] / 4;  // 4096*1024 tokens
    const int blocks = 2048;           // grid-stride covers all tile pairs

    moe_wmma_kernel<<<blocks, BLOCK, 0, stream>>>(
        x, w1, b1, gate_w, expert_w, expert_b, res_w, res_b,
        comb_w, comb_b, w2, b2, (float*)d_out, ntok);
}